// GMMConv_67508295958895
// MI455X (gfx1250) — compile-verified
//
#include <hip/hip_runtime.h>

// GMMConv (MoNet) forward for MI455X / gfx1250 (wave32).
//
// N=100000, DEG=16, E=1.6M, D=2, K=4, IN_F=128, OUT_F=32.
//
// Stage 1: node_feat = feat @ fc_w^T via v_wmma_f32_16x16x4_f32 (exact fp32),
//          rows stored transposed as [f*4+k] so stage 2 gathers one float4.
// Stage 2: fused gaussian weights + gather + segment-sum + bias.
//          node_feat (51.2MB) is L2-resident (192MB L2) so the 819MB gather
//          stays on-chip; HBM floor ~130MB @ 23.3TB/s.

typedef __attribute__((ext_vector_type(2))) float v2f;
typedef __attribute__((ext_vector_type(8))) float v8f;

#define IN_F 128
#define OUT_F 32

__device__ __forceinline__ float lane_bcast(float x, int l) {
    return __int_as_float(__builtin_amdgcn_readlane(__float_as_int(x), l));
}

// ---------------------------------------------------------------------------
// Stage 1: [N,128] x [128,128]^T GEMM, WMMA f32 16x16x4, fully unrolled K.
// One wave = one 16(M) x 16(N) tile; 8 waves/block = one 16-row M-tile.
// A (16x4 f32): lanes 0-15 M=lane, V0/V1 = K0/K1; lanes 16-31 V0/V1 = K2/K3.
// B (4x16 f32): mirrored K striping, lane = N.
// C/D: VGPR v -> M = v + 8*(lane>=16), N = lane&15.
// ---------------------------------------------------------------------------
__global__ void __launch_bounds__(256)
gmm_gemm_wmma(const float* __restrict__ feat,   // [N, IN_F]
              const float* __restrict__ fcw,    // [K*OUT_F, IN_F]
              float* __restrict__ nft,          // [N, IN_F] transposed-in-row
              int n)
{
    const int lane  = threadIdx.x & 31;
    const int wave  = threadIdx.x >> 5;
    const int mBase = blockIdx.x << 4;
    const int nBase = wave << 4;

    const int m  = lane & 15;
    const int kh = (lane >> 4) << 1;            // 0 or 2

    int arow = mBase + m;
    if (arow > n - 1) arow = n - 1;             // clamp: EXEC stays all-1
    const float* __restrict__ aptr = feat + (size_t)arow * IN_F + kh;
    const float* __restrict__ bptr = fcw  + (size_t)(nBase + m) * IN_F + kh;

    v8f c = {};
#pragma unroll
    for (int k0 = 0; k0 < IN_F; k0 += 4) {
        v2f a = *(const v2f*)(aptr + k0);       // 8B-aligned (k0+kh even)
        v2f b = *(const v2f*)(bptr + k0);
        c = __builtin_amdgcn_wmma_f32_16x16x4_f32(
                false, a, false, b, (short)0, c, false, false);
    }

    // In-row transpose on store: column n = k*32 + f  ->  index f*4 + k.
    const int nCol = nBase + (lane & 15);
    const int tIdx = ((nCol & 31) << 2) + (nCol >> 5);
    const int rowB = mBase + ((lane >> 4) << 3);
    float* __restrict__ dst = nft + (size_t)rowB * IN_F + tIdx;

    if (mBase + 16 <= n) {                      // fast path: no guards
#pragma unroll
        for (int v = 0; v < 8; ++v)
            dst[(size_t)v * IN_F] = c[v];
    } else {
#pragma unroll
        for (int v = 0; v < 8; ++v)
            if (rowB + v < n)
                dst[(size_t)v * IN_F] = c[v];
    }
}

// ---------------------------------------------------------------------------
// Stage 2: fused per-edge gaussian weights + gather + segment-sum + bias.
// One wave per dst node; lane = output channel f (OUT_F == 32 == wave width).
// rs/deg forced wave-uniform -> colind loads scalarize (SMEM), gathers use
// SADDR form; weights broadcast via v_readlane (no LDS, no barrier).
// ---------------------------------------------------------------------------
__global__ void __launch_bounds__(256)
gmm_edge_fused(const int*   __restrict__ rowptr,   // [N+1]
               const int*   __restrict__ colind,   // [E]
               const int*   __restrict__ permute,  // [E]
               const float* __restrict__ pseudo,   // [E, 2]
               const float* __restrict__ mu,       // [K, 2]
               const float* __restrict__ isig,     // [K, 2]
               const float* __restrict__ bias,     // [OUT_F]
               const float* __restrict__ nft,      // [N, IN_F] transposed rows
               float*       __restrict__ out,      // [N, OUT_F]
               int n)
{
    const int lane = threadIdx.x & 31;
    const int node = (blockIdx.x << 3) + (threadIdx.x >> 5);
    const bool active = (node < n);             // uniform per wave

    int rs = 0, deg = 0;
    if (active) {
        rs  = rowptr[node];
        deg = rowptr[node + 1] - rs;
        if (deg > 16) deg = 16;                 // weight-broadcast capacity
    }
    rs  = __builtin_amdgcn_readfirstlane(rs);   // wave-uniform by construction
    deg = __builtin_amdgcn_readfirstlane(deg);

    // Each lane computes 2 of the 64 weights: idx = 2*lane+t, el=idx>>2, k=idx&3.
    float wa = 0.0f, wb = 0.0f;
#pragma unroll
    for (int t = 0; t < 2; ++t) {
        const int idx = (lane << 1) + t;
        const int el  = idx >> 2;
        const int k   = idx & 3;
        float w = 0.0f;
        if (el < deg) {
            const int pe   = permute[rs + el];
            const float p0 = pseudo[2 * pe];
            const float p1 = pseudo[2 * pe + 1];
            const float d0 = (p0 - mu[2 * k])     * isig[2 * k];
            const float d1 = (p1 - mu[2 * k + 1]) * isig[2 * k + 1];
            w = __expf(-0.5f * (d0 * d0 + d1 * d1));
        }
        if (t == 0) wa = w; else wb = w;
    }

    float acc = active ? bias[lane] : 0.0f;

    auto edge = [&](int el) {
        const int src = colind[rs + el];        // uniform -> scalar load
        const float4 v = *((const float4*)(nft + (size_t)src * IN_F) + lane);
        const float w0 = lane_bcast(wa, 2 * el);
        const float w1 = lane_bcast(wb, 2 * el);
        const float w2 = lane_bcast(wa, 2 * el + 1);
        const float w3 = lane_bcast(wb, 2 * el + 1);
        acc = fmaf(w0, v.x, acc);
        acc = fmaf(w1, v.y, acc);
        acc = fmaf(w2, v.z, acc);
        acc = fmaf(w3, v.w, acc);
    };

    if (deg == 16) {                            // common case: full MLP unroll
#pragma unroll
        for (int el = 0; el < 16; ++el) edge(el);
    } else {
        for (int el = 0; el < deg; ++el) edge(el);
    }

    if (active)
        out[(size_t)node * OUT_F + lane] = acc;
}

// ---------------------------------------------------------------------------
extern "C" void kernel_launch(void* const* d_in, const int* in_sizes, int n_in,
                              void* d_out, int out_size, void* d_ws, size_t ws_size,
                              hipStream_t stream)
{
    const int*   rowptr  = (const int*)  d_in[0];
    const int*   colind  = (const int*)  d_in[1];
    // d_in[2] = colptr, d_in[3] = rowind: backward-only mirrors, unused in fwd
    const int*   permute = (const int*)  d_in[4];
    const float* feat    = (const float*)d_in[5];
    const float* pseudo  = (const float*)d_in[6];
    const float* fcw     = (const float*)d_in[7];
    const float* mu      = (const float*)d_in[8];
    const float* isig    = (const float*)d_in[9];
    const float* bias    = (const float*)d_in[10];
    float*       out     = (float*)d_out;

    const int n = in_sizes[0] - 1;              // rowptr has N+1 entries
    float* nft = (float*)d_ws;                  // [N, 128] = 51.2 MB scratch

    gmm_gemm_wmma<<<(n + 15) >> 4, 256, 0, stream>>>(feat, fcw, nft, n);
    gmm_edge_fused<<<(n + 7) >> 3, 256, 0, stream>>>(rowptr, colind, permute,
                                                     pseudo, mu, isig, bias,
                                                     nft, out, n);
}